// DeepGpcmModel_72249939853468
// MI455X (gfx1250) — compile-verified
//
#include <hip/hip_runtime.h>
#include <hip/hip_bf16.h>
#include <math.h>

typedef __attribute__((ext_vector_type(2))) float v2f;
typedef __attribute__((ext_vector_type(8))) float v8f;
typedef int v4i_ __attribute__((vector_size(16)));   // matches builtin param type

constexpr int Bn = 256, Sn = 200, Tn = Bn * Sn;   // tokens
constexpr int NQ = 400, Kc = 4, Mm = 50, KD = 50, VD = 200, FD = 50;
constexpr int FW = 256;                            // padded feature row stride
constexpr int KP = 252;                            // padded K for summary GEMM

__device__ __forceinline__ float sigmoidf_(float x) { return 1.f / (1.f + __expf(-x)); }

// --- CDNA5 async global->LDS copy (ASYNCcnt path), with sync fallback -------
#if defined(__has_builtin)
#if __has_builtin(__builtin_amdgcn_global_load_async_to_lds_b128) && \
    __has_builtin(__builtin_amdgcn_s_wait_asynccnt)
#define HAVE_ASYNC_LDS 1
#endif
#endif

#ifdef HAVE_ASYNC_LDS
__device__ __forceinline__ void async_copy16(const float* g, float* l) {
  // 16B global -> LDS, tracked by ASYNCcnt
  __builtin_amdgcn_global_load_async_to_lds_b128(
      (__attribute__((address_space(1))) v4i_*)g,
      (__attribute__((address_space(3))) v4i_*)l, 0, 0);
}
__device__ __forceinline__ void async_drain() {
  __builtin_amdgcn_s_wait_asynccnt(0);
}
#else
__device__ __forceinline__ void async_copy16(const float* g, float* l) {
  l[0] = g[0]; l[1] = g[1]; l[2] = g[2]; l[3] = g[3];
}
__device__ __forceinline__ void async_drain() {}
#endif

// ---------------------------------------------------------------------------
// Kernel A: per-token embeddings, value_emb, query_key, corr = softmax(qk@keyT)
// Also writes q_emb into feat[t][200..249] and zeros feat[t][250..255].
// ---------------------------------------------------------------------------
__global__ void kA(const int* __restrict__ q_data, const int* __restrict__ r_data,
                   const float* __restrict__ q_tab, const float* __restrict__ key_mem,
                   const float* __restrict__ W_qk, const float* __restrict__ b_qk,
                   const float* __restrict__ W_v3, const float* __restrict__ b_v,
                   float* __restrict__ feat, float* __restrict__ vemb,
                   float* __restrict__ corr) {
  __shared__ float s_qe[KD], s_qk[KD], s_sc[Mm];
  const int t = blockIdx.x, tid = threadIdx.x;
  const int q = q_data[t], r = r_data[t];
  const float qmask = (q > 0) ? 1.f : 0.f;
  int idx = q - 1; idx = idx < 0 ? 0 : (idx > NQ - 1 ? NQ - 1 : idx);

  if (tid < KD) {
    float v = q_tab[q * KD + tid];
    s_qe[tid] = v;
    feat[(size_t)t * FW + VD + tid] = v;            // q_emb section of feat
  }
  if (tid >= VD + KD && tid < FW) {
    feat[(size_t)t * FW + tid] = 0.f;               // cols 250..255 = 0
  }
  if (tid < VD) {
    float acc = b_v[tid];
#pragma unroll
    for (int k = 0; k < Kc; ++k) {
      float w = 1.f - fabsf((float)k - (float)r) / (float)(Kc - 1);
      w = w > 0.f ? w : 0.f;
      acc += w * qmask * W_v3[((size_t)k * NQ + idx) * VD + tid];
    }
    vemb[(size_t)t * VD + tid] = acc;
  }
  __syncthreads();
  if (tid < KD) {
    float acc = b_qk[tid];
    for (int j = 0; j < KD; ++j) acc += s_qe[j] * W_qk[j * KD + tid];
    s_qk[tid] = tanhf(acc);
  }
  __syncthreads();
  if (tid < Mm) {
    float acc = 0.f;
    for (int j = 0; j < KD; ++j) acc += s_qk[j] * key_mem[tid * KD + j];
    s_sc[tid] = acc;
  }
  __syncthreads();
  if (tid < Mm) {
    float mx = -1e30f;
    for (int m = 0; m < Mm; ++m) mx = fmaxf(mx, s_sc[m]);
    float sum = 0.f;
    for (int m = 0; m < Mm; ++m) sum += __expf(s_sc[m] - mx);
    corr[(size_t)t * Mm + tid] = __expf(s_sc[tid] - mx) / sum;
  }
}

// ---------------------------------------------------------------------------
// Kernel B: erase = sigmoid(vemb@W_e + b_e), add = tanh(vemb@W_a + b_a)
// [T,200]x[200,200] via V_WMMA_F32_16X16X4_F32. B tiles (200x16 slices of
// W_e/W_a) are async-staged into LDS once per block, zero-padded; the K loop
// is branch-free: global b64 (A) + LDS reads (B) + 2 WMMAs.
// ---------------------------------------------------------------------------
__global__ void kB(const float* __restrict__ vemb, const float* __restrict__ We,
                   const float* __restrict__ Wa, const float* __restrict__ be,
                   const float* __restrict__ ba, float* __restrict__ er,
                   float* __restrict__ ad) {
  __shared__ float s_We[VD * 16];
  __shared__ float s_Wa[VD * 16];
  const int lane = threadIdx.x & 31;
  const int wave = threadIdx.x >> 5;
  const int half = lane >> 4;
  const int l16  = lane & 15;
  const int row0 = (blockIdx.x * 8 + wave) * 16;
  const int col0 = blockIdx.y * 16;

  // Stage W_e / W_a column slices -> LDS (16B quads; async where fully valid).
  for (int i = threadIdx.x; i < VD * 4; i += 256) {   // 800 quad tasks
    const int rr = i >> 2, qq = i & 3;
    const int c0 = col0 + qq * 4;
    float* le = &s_We[rr * 16 + qq * 4];
    float* la = &s_Wa[rr * 16 + qq * 4];
    if (c0 + 3 < VD) {
      async_copy16(We + (size_t)rr * VD + c0, le);
      async_copy16(Wa + (size_t)rr * VD + c0, la);
    } else {
#pragma unroll
      for (int j = 0; j < 4; ++j) {
        const int c = c0 + j;
        le[j] = (c < VD) ? We[(size_t)rr * VD + c] : 0.f;
        la[j] = (c < VD) ? Wa[(size_t)rr * VD + c] : 0.f;
      }
    }
  }
  async_drain();
  __syncthreads();

  v8f accE = {}; v8f accA = {};
  for (int kk = 0; kk < VD / 4; ++kk) {
    const int k0 = kk * 4 + 2 * half;                 // K of lane element 0
    v2f a = *(const v2f*)(vemb + (size_t)(row0 + l16) * VD + k0);
    v2f bE, bA;
    bE.x = s_We[(k0    ) * 16 + l16];
    bE.y = s_We[(k0 + 1) * 16 + l16];
    bA.x = s_Wa[(k0    ) * 16 + l16];
    bA.y = s_Wa[(k0 + 1) * 16 + l16];
    accE = __builtin_amdgcn_wmma_f32_16x16x4_f32(false, a, false, bE, (short)0, accE, false, false);
    accA = __builtin_amdgcn_wmma_f32_16x16x4_f32(false, a, false, bA, (short)0, accA, false, false);
  }

  const int n = col0 + l16;
  if (n < VD) {
    const float biE = be[n], biA = ba[n];
#pragma unroll
    for (int i = 0; i < 8; ++i) {
      const int row = row0 + i + 8 * half;            // D layout: VGPR i -> M = i + 8*half
      er[(size_t)row * VD + n] = sigmoidf_(accE[i] + biE);
      ad[(size_t)row * VD + n] = tanhf(accA[i] + biA);
    }
  }
}

// ---------------------------------------------------------------------------
// Kernel C: sequential DKVMN scan. One block per batch, mem (50x200 f32=40KB)
// in LDS; thread v owns column v. reads are written into feat[t][0..199].
// ---------------------------------------------------------------------------
__global__ void kC(const float* __restrict__ corr, const float* __restrict__ er,
                   const float* __restrict__ ad, const float* __restrict__ init_mem,
                   float* __restrict__ feat) {
  __shared__ float s_mem[Mm * VD];
  __shared__ float s_w[Mm];
  const int b = blockIdx.x, tid = threadIdx.x;
  for (int i = tid; i < Mm * VD; i += blockDim.x) s_mem[i] = init_mem[i];
  __syncthreads();
  for (int s = 0; s < Sn; ++s) {
    const int t = b * Sn + s;
    if (tid < Mm) s_w[tid] = corr[(size_t)t * Mm + tid];
    __syncthreads();
    if (tid < VD) {
      const float e = er[(size_t)t * VD + tid];
      const float a = ad[(size_t)t * VD + tid];
      float rd = 0.f;
#pragma unroll 5
      for (int m = 0; m < Mm; ++m) {
        const float w  = s_w[m];
        const float mv = s_mem[m * VD + tid];
        rd += w * mv;
        s_mem[m * VD + tid] = mv * (1.f - w * e) + w * a;
      }
      feat[(size_t)t * FW + tid] = rd;
    }
    __syncthreads();
  }
}

// ---------------------------------------------------------------------------
// Kernel D: summary = tanh(feat @ W_sum + b_sum), feat = [reads|q_emb|0] with
// row stride 256; K padded to 252; W_sum staged zero-padded into LDS.
// ---------------------------------------------------------------------------
__global__ void kD(const float* __restrict__ feat, const float* __restrict__ W_sum,
                   const float* __restrict__ b_sum, float* __restrict__ summary) {
  __shared__ float s_Ws[KP * 16];
  const int lane = threadIdx.x & 31;
  const int wave = threadIdx.x >> 5;
  const int half = lane >> 4;
  const int l16  = lane & 15;
  const int row0 = (blockIdx.x * 8 + wave) * 16;
  const int col0 = blockIdx.y * 16;

  for (int i = threadIdx.x; i < KP * 16; i += 256) {
    const int k = i >> 4, c = i & 15;
    const int n2 = col0 + c;
    s_Ws[i] = (k < VD + KD && n2 < FD) ? W_sum[(size_t)k * FD + n2] : 0.f;
  }
  __syncthreads();

  v8f acc = {};
  for (int kk = 0; kk < KP / 4; ++kk) {
    const int k0 = kk * 4 + 2 * half;
    v2f a = *(const v2f*)(feat + (size_t)(row0 + l16) * FW + k0);
    v2f bv;
    bv.x = s_Ws[(k0    ) * 16 + l16];
    bv.y = s_Ws[(k0 + 1) * 16 + l16];
    acc = __builtin_amdgcn_wmma_f32_16x16x4_f32(false, a, false, bv, (short)0, acc, false, false);
  }

  const int n = col0 + l16;
  if (n < FD) {
    const float bi = b_sum[n];
#pragma unroll
    for (int i = 0; i < 8; ++i) {
      const int row = row0 + i + 8 * half;
      summary[(size_t)row * FD + n] = tanhf(acc[i] + bi);
    }
  }
}

// ---------------------------------------------------------------------------
// Kernel E: IRT head, one thread per token. Streams summary / q_emb once.
// ---------------------------------------------------------------------------
__global__ void kE(const float* __restrict__ summary, const float* __restrict__ feat,
                   const float* __restrict__ W_ab, const float* __restrict__ b_ab,
                   const float* __restrict__ W_th, const float* __restrict__ b_th,
                   const float* __restrict__ W_disc, const float* __restrict__ b_disc,
                   float* __restrict__ out) {
  const int t = blockIdx.x * blockDim.x + threadIdx.x;
  if (t >= Tn) return;

  float th = b_ab[0];
  float da = b_disc[0];
  for (int i = 0; i < FD; ++i) {
    const float s = summary[(size_t)t * FD + i];
    th += s * W_ab[i];
    da += s * W_disc[i];
  }
  th *= 3.0f;                                   // ABILITY_SCALE

  float bb0 = b_th[0], bb1 = b_th[1], bb2 = b_th[2];
  for (int i = 0; i < KD; ++i) {
    const float qv = feat[(size_t)t * FW + VD + i];   // q_emb section
    da  += qv * W_disc[FD + i];
    bb0 += qv * W_th[i * (Kc - 1) + 0];
    bb1 += qv * W_th[i * (Kc - 1) + 1];
    bb2 += qv * W_th[i * (Kc - 1) + 2];
  }
  const float alpha = (da > 20.f) ? da : log1pf(__expf(da));

  float logits[Kc];
  logits[0] = 0.f;
  float cum = alpha * (th - tanhf(bb0)); logits[1] = cum;
  cum += alpha * (th - tanhf(bb1));      logits[2] = cum;
  cum += alpha * (th - tanhf(bb2));      logits[3] = cum;

  float mx = logits[0];
  for (int j = 1; j < Kc; ++j) mx = fmaxf(mx, logits[j]);
  float sum = 0.f;
  for (int j = 0; j < Kc; ++j) { logits[j] = __expf(logits[j] - mx); sum += logits[j]; }
  for (int j = 0; j < Kc; ++j) out[(size_t)t * Kc + j] = logits[j] / sum;
}

extern "C" void kernel_launch(void* const* d_in, const int* in_sizes, int n_in,
                              void* d_out, int out_size, void* d_ws, size_t ws_size,
                              hipStream_t stream) {
  (void)in_sizes; (void)n_in; (void)out_size; (void)ws_size;
  const int*   q_data  = (const int*)  d_in[0];
  const int*   r_data  = (const int*)  d_in[1];
  const float* q_tab   = (const float*)d_in[2];
  const float* key_mem = (const float*)d_in[3];
  const float* init_mem= (const float*)d_in[4];
  const float* W_qk    = (const float*)d_in[5];
  const float* b_qk    = (const float*)d_in[6];
  const float* W_v3    = (const float*)d_in[7];
  const float* b_v     = (const float*)d_in[8];
  const float* W_e     = (const float*)d_in[9];
  const float* b_e     = (const float*)d_in[10];
  const float* W_a     = (const float*)d_in[11];
  const float* b_a     = (const float*)d_in[12];
  const float* W_sum   = (const float*)d_in[13];
  const float* b_sum   = (const float*)d_in[14];
  const float* W_ab    = (const float*)d_in[15];
  const float* b_ab    = (const float*)d_in[16];
  const float* W_th    = (const float*)d_in[17];
  const float* b_th    = (const float*)d_in[18];
  const float* W_disc  = (const float*)d_in[19];
  const float* b_disc  = (const float*)d_in[20];
  float* out = (float*)d_out;

  // workspace layout (floats). summary aliases vemb (vemb dead after kB).
  float* w     = (float*)d_ws;
  float* vemb  = w;                                  // T*VD
  float* corr  = vemb + (size_t)Tn * VD;             // T*Mm
  float* erase = corr + (size_t)Tn * Mm;             // T*VD
  float* add   = erase + (size_t)Tn * VD;            // T*VD
  float* feat  = add   + (size_t)Tn * VD;            // T*FW (reads|q_emb|0)
  float* summary = vemb;                             // alias, T*FD

  // A: per-token features (+ q_emb into feat, pad cols zeroed)
  kA<<<Tn, 256, 0, stream>>>(q_data, r_data, q_tab, key_mem, W_qk, b_qk, W_v3, b_v,
                             feat, vemb, corr);
  // B: WMMA GEMMs for erase/add (M tiles: 3200 -> 400 blocks x 8 waves; N tiles: 13)
  kB<<<dim3(Tn / 16 / 8, (VD + 15) / 16), 256, 0, stream>>>(vemb, W_e, W_a, b_e, b_a,
                                                            erase, add);
  // C: sequential scan, one block per batch; reads -> feat[:,0:200]
  kC<<<Bn, 256, 0, stream>>>(corr, erase, add, init_mem, feat);
  // D: WMMA GEMM for summary (N tiles: 4)
  kD<<<dim3(Tn / 16 / 8, (FD + 15) / 16), 256, 0, stream>>>(feat, W_sum, b_sum, summary);
  // E: IRT head -> probs
  kE<<<(Tn + 255) / 256, 256, 0, stream>>>(summary, feat, W_ab, b_ab, W_th, b_th,
                                           W_disc, b_disc, out);
}